// GMP_27075473834661
// MI455X (gfx1250) — compile-verified
//
#include <hip/hip_runtime.h>
#include <math.h>

// GMP (generalized memory polynomial) on gfx1250.
// S[t,m] = sum_u A_u[t,k] * B_u[k,m] via V_WMMA_F32_16X16X4_F32, u = l+m in [0,20].

typedef __attribute__((ext_vector_type(2))) float v2f;
typedef __attribute__((ext_vector_type(4))) float v4f;
typedef __attribute__((ext_vector_type(8))) float v8f;

#define MTAP   11
#define KPOW   4            // powers 1..4 == WMMA K dimension
#define TT     4096
#define BBATCH 32
#define TILE_T 128          // t-samples per block
#define WAVE_T 16           // t-samples per wave (WMMA M)
#define NWAVE  8            // 256 threads / wave32
#define HALO   20           // max lookback: (M-1)+(M-1) = 20
#define NS     (TILE_T + HALO)   // 148 staged samples
#define NU     21           // u = l+m in [0,20]

__global__ __launch_bounds__(256)
void gmp_wmma_kernel(const float* __restrict__ x,     // (B,T,2)
                     const float* __restrict__ Wt,    // (1, 11 + 4*11*11)
                     float* __restrict__ out)         // (B,T,2)
{
    __shared__ v4f   s_pows[NS];          // {a, a^2, a^3, a^4} per sample
    __shared__ v2f   s_x[NS];             // {re, im} per sample
    __shared__ v2f   s_B[NU * 32];        // B fragment per u per lane
    __shared__ float s_S[NWAVE][WAVE_T][17];  // stride 17: avoid bank conflict

    const int tid    = threadIdx.x;
    const int lane   = tid & 31;
    const int wave   = tid >> 5;
    const int b      = blockIdx.x / (TT / TILE_T);
    const int t_base = (blockIdx.x % (TT / TILE_T)) * TILE_T;

    // ---- stage input window + amplitude powers into LDS ----
    for (int i = tid; i < NS; i += 256) {
        const int t = t_base - HALO + i;
        float re = 0.f, im = 0.f;
        if (t >= 0) {
            re = x[((size_t)b * TT + t) * 2 + 0];
            im = x[((size_t)b * TT + t) * 2 + 1];
        }
        const float a2 = re * re + im * im;
        const float a1 = sqrtf(a2);
        v4f p; p.x = a1; p.y = a2; p.z = a1 * a2; p.w = a2 * a2;
        s_pows[i] = p;
        v2f xv; xv.x = re; xv.y = im;
        s_x[i] = xv;
    }

    // ---- stage B_u fragments: B_u[k,n] = W1[k, u-n, n] ----
    // B 4x16 layout: lanes 0-15 hold col n=lane with K=0,1 in the two dwords;
    // lanes 16-31 hold col n=lane-16 with K=2,3 (mirrors the A 16x4 layout).
    const float* W1 = Wt + MTAP;  // [k][l][m], strides 121/11/1
    for (int idx = tid; idx < NU * 32; idx += 256) {
        const int u  = idx >> 5;
        const int L  = idx & 31;
        const int n  = L & 15;
        const int k0 = (L >> 4) * 2;
        const int l  = u - n;
        v2f bv; bv.x = 0.f; bv.y = 0.f;
        if (n < MTAP && l >= 0 && l < MTAP) {
            bv.x = W1[(k0 * MTAP + l) * MTAP + n];
            bv.y = W1[((k0 + 1) * MTAP + l) * MTAP + n];
        }
        s_B[idx] = bv;
    }
    __syncthreads();

    // ---- WMMA accumulation over u ----
    // A 16x4 f32 layout: lanes 0-15: M=lane, K=0,1; lanes 16-31: M=lane-16, K=2,3.
    const int row   = lane & 15;
    const int hi    = lane >> 4;          // 0 -> powers {1,2}, 1 -> powers {3,4}
    const int abase = wave * WAVE_T + row;
    v8f acc = {};
    #pragma unroll
    for (int u = 0; u < NU; ++u) {
        const float* pp = (const float*)&s_pows[abase + u];
        v2f a; a.x = pp[2 * hi + 0]; a.y = pp[2 * hi + 1];
        const v2f bv = s_B[u * 32 + lane];
        acc = __builtin_amdgcn_wmma_f32_16x16x4_f32(
            false, a, false, bv, (short)0, acc, false, false);
    }

    // ---- spill 16x16 S tile: VGPR r holds M=r (lanes 0-15) / M=r+8 (16-31) ----
    #pragma unroll
    for (int r = 0; r < 8; ++r) {
        s_S[wave][r + 8 * hi][row] = acc[r];
    }
    __syncthreads();

    // ---- final combine: out[t] = sum_m xc[t+m-10] * (W0[m] + S[t,m]) ----
    if (lane < WAVE_T) {
        const int trow = lane;
        const int t    = t_base + wave * WAVE_T + trow;
        float o_re = 0.f, o_im = 0.f;
        #pragma unroll
        for (int n = 0; n < MTAP; ++n) {
            const float c  = Wt[n] + s_S[wave][trow][n];
            const v2f  xv  = s_x[wave * WAVE_T + trow + n + (HALO - MTAP + 1)];
            o_re += xv.x * c;
            o_im += xv.y * c;
        }
        const size_t o = ((size_t)b * TT + t) * 2;
        out[o + 0] = o_re;
        out[o + 1] = o_im;
    }
}

extern "C" void kernel_launch(void* const* d_in, const int* in_sizes, int n_in,
                              void* d_out, int out_size, void* d_ws, size_t ws_size,
                              hipStream_t stream) {
    const float* x  = (const float*)d_in[0];   // (32, 4096, 2) f32
    // d_in[1] = h_0 (unused by the reference computation)
    const float* Wt = (const float*)d_in[2];   // (1, 495) f32
    float* out = (float*)d_out;                // (32, 4096, 2) f32

    const int blocks = BBATCH * (TT / TILE_T); // 1024
    gmp_wmma_kernel<<<blocks, 256, 0, stream>>>(x, Wt, out);
}